// Encoder_34943853920917
// MI455X (gfx1250) — compile-verified
//
#include <hip/hip_runtime.h>
#include <hip/hip_bf16.h>
#include <math.h>

// GRU encoder for MI455X (gfx1250), compile-only tuning.
//
// Phase 1: gi = src @ w_ih^T + b_ih  via native-fp32 WMMA (v_wmma_f32_16x16x4_f32).
// Phase 2: persistent 256-WG scan kernel; each WGP keeps its 24 rows of w_hh
//          (196 KB) + staged h (8 KB) resident in the 320 KB LDS for all 4096
//          steps; grid-wide sense-reversing barrier per step.
//
// Workspace layout (requires ~100.7 MB):
//   [0, GI_BYTES)            : gi, fp32 [4096, 6144]
//   [GI_BYTES, +16KB)        : h double buffer, fp32 [2, 2048]
//   [BAR_OFF, +8B)           : grid barrier {count, generation}

#define L_SEQ 4096
#define I_IN  2048
#define H_HID 2048
#define G3    (3 * H_HID)      // 6144

typedef float v2f __attribute__((ext_vector_type(2)));
typedef float v8f __attribute__((ext_vector_type(8)));

static constexpr size_t GI_BYTES = (size_t)L_SEQ * G3 * sizeof(float);   // 100,663,296
static constexpr size_t HBUF_OFF = GI_BYTES;
static constexpr size_t BAR_OFF  = HBUF_OFF + (size_t)2 * H_HID * sizeof(float);

// ---------------------------------------------------------------------------
// Init: zero barrier words, seed h double-buffer with h0. Re-run every launch
// so graph replay is deterministic (d_ws is poisoned once, never re-poisoned).
// ---------------------------------------------------------------------------
__global__ void gru_init_kernel(const float* __restrict__ h0,
                                float* __restrict__ hbufs,
                                unsigned* __restrict__ bar) {
    int i = blockIdx.x * blockDim.x + threadIdx.x;
    if (i < H_HID) {
        hbufs[i] = h0[i];          // buffer 0: h at t=0
        hbufs[H_HID + i] = 0.0f;   // buffer 1: scratch
    }
    if (i < 2) bar[i] = 0u;
}

// ---------------------------------------------------------------------------
// Phase 1: gi[m, n] = sum_k src[m, k] * w_ih[n, k] + b_ih[n]
// One wave -> 16(M) x 64(N) strip, 4 v8f accumulators, K stepped by 4 with
// v_wmma_f32_16x16x4_f32. Fragment layouts per CDNA5 ISA 7.12.2:
//   A 16x4 f32: lanes 0-15 = M rows, VGPR{0,1} = K{0,1} (lanes 16-31: K{2,3})
//   B 4x16 f32: lanes 0-15 = N cols, VGPR{0,1} = K{0,1} (lanes 16-31: K{2,3})
// Both src and w_ih are K-contiguous, so each lane does one 8-byte load per
// fragment; lines are reused across 4 consecutive k-steps and both operands
// are L2-resident (32 MB + 48 MB << 192 MB L2).
// ---------------------------------------------------------------------------
__global__ void __launch_bounds__(256)
gru_gemm_wmma(const float* __restrict__ src,
              const float* __restrict__ w_ih,
              const float* __restrict__ b_ih,
              float* __restrict__ gi) {
    const int lane = threadIdx.x & 31;
    const int wave = threadIdx.x >> 5;

    const int m0 = blockIdx.y << 4;                    // 16-row M tile
    const int n0 = (blockIdx.x << 9) + (wave << 6);    // 64-col N strip per wave

    const int row  = lane & 15;            // M row (A) / N col (B) within tile
    const int koff = (lane >> 4) << 1;     // lanes 16-31 hold K+2,K+3

    const float* aptr = src  + (size_t)(m0 + row) * I_IN + koff;
    const float* bptr = w_ih + (size_t)(n0 + row) * I_IN + koff;

    v8f acc[4];
#pragma unroll
    for (int nt = 0; nt < 4; ++nt)
        acc[nt] = (v8f){0.f, 0.f, 0.f, 0.f, 0.f, 0.f, 0.f, 0.f};

    for (int k0 = 0; k0 < I_IN; k0 += 4) {
        v2f a = *(const v2f*)(aptr + k0);
#pragma unroll
        for (int nt = 0; nt < 4; ++nt) {
            v2f b = *(const v2f*)(bptr + ((size_t)nt << 15) + k0);  // nt*16*2048
            acc[nt] = __builtin_amdgcn_wmma_f32_16x16x4_f32(
                false, a, false, b, (short)0, acc[nt], false, false);
        }
    }

    // C/D layout: VGPR v, lanes 0-15 -> M=m0+v, lanes 16-31 -> M=m0+v+8.
#pragma unroll
    for (int nt = 0; nt < 4; ++nt) {
        const int ncol = n0 + (nt << 4) + (lane & 15);
        const float bias = b_ih[ncol];
#pragma unroll
        for (int v = 0; v < 8; ++v) {
            const int mrow = m0 + v + ((lane >> 4) << 3);
            gi[(size_t)mrow * G3 + ncol] = acc[nt][v] + bias;
        }
    }
}

// ---------------------------------------------------------------------------
// Grid-wide sense-reversing barrier (persistent kernel, 256 resident WGs).
// ---------------------------------------------------------------------------
__device__ __forceinline__ void grid_barrier(unsigned* cnt, unsigned* gen,
                                             unsigned total) {
    __syncthreads();
    if (threadIdx.x == 0) {
        unsigned g = __atomic_load_n(gen, __ATOMIC_ACQUIRE);
        if (atomicAdd(cnt, 1u) == total - 1u) {
            __atomic_store_n(cnt, 0u, __ATOMIC_RELAXED);
            __threadfence();
            __atomic_fetch_add(gen, 1u, __ATOMIC_ACQ_REL);
        } else {
            while (__atomic_load_n(gen, __ATOMIC_ACQUIRE) == g)
                __builtin_amdgcn_s_sleep(1);
        }
    }
    __syncthreads();
}

// ---------------------------------------------------------------------------
// Phase 2: persistent GRU scan. 256 WGs x 256 threads (8 waves).
// WG wg owns hidden units [wg*8, wg*8+8); wave w owns unit u = wg*8 + w and
// its three w_hh rows (r, z, n), pinned in LDS. Per step: stage h into LDS,
// 3 dot products per wave (fp32 FMA, stride-32 -> bank-conflict-free),
// wave32 shuffle reduction, gate math on lane 0, double-buffered h write,
// grid barrier.
// LDS: 24*2048*4 (weights) + 2048*4 (h) = 204,800 B <= 320 KB/WGP.
// ---------------------------------------------------------------------------
__global__ void __launch_bounds__(256, 1)
gru_scan(const float* __restrict__ gi,
         const float* __restrict__ w_hh,
         const float* __restrict__ b_hh,
         float* __restrict__ hbufs,
         unsigned* __restrict__ bar,
         float* __restrict__ out) {
    extern __shared__ float smem[];
    float* wlds = smem;                 // [24][2048]
    float* hlds = smem + 24 * 2048;     // [2048]

    const int tid  = threadIdx.x;
    const int lane = tid & 31;
    const int wave = tid >> 5;
    const int j0   = blockIdx.x << 3;   // first hidden unit of this WG
    const int u    = j0 + wave;         // this wave's hidden unit

    // One-time fill of this WGP's w_hh slice into LDS (coalesced).
    for (int idx = tid; idx < 24 * 2048; idx += 256) {
        const int lrow = idx >> 11;
        const int col  = idx & 2047;
        const int lw   = lrow / 3;          // local hidden unit
        const int g    = lrow - 3 * lw;     // gate: 0=r, 1=z, 2=n
        const int grow = j0 + lw + g * H_HID;
        wlds[idx] = w_hh[(size_t)grow * H_HID + col];
    }

    const float bh_r = b_hh[u];
    const float bh_z = b_hh[u + H_HID];
    const float bh_n = b_hh[u + 2 * H_HID];
    const float* wr = wlds + ((wave * 3 + 0) << 11);
    const float* wz = wlds + ((wave * 3 + 1) << 11);
    const float* wn = wlds + ((wave * 3 + 2) << 11);

    unsigned* cnt = bar;
    unsigned* gen = bar + 1;

    float hnew = 0.0f;

    for (int t = 0; t < L_SEQ; ++t) {
        const float* hcur = hbufs + ((t & 1) << 11);
        float*       hnxt = hbufs + (((t + 1) & 1) << 11);

        // Stage h_t into LDS (previous grid_barrier's trailing __syncthreads
        // guarantees last step's readers are done with hlds).
        for (int i = tid; i < H_HID; i += 256) hlds[i] = hcur[i];
        __syncthreads();

        float a0 = 0.f, a1 = 0.f, a2 = 0.f;
        for (int e = lane; e < H_HID; e += 32) {
            const float hv = hlds[e];
            a0 = fmaf(wr[e], hv, a0);
            a1 = fmaf(wz[e], hv, a1);
            a2 = fmaf(wn[e], hv, a2);
        }
#pragma unroll
        for (int off = 16; off > 0; off >>= 1) {
            a0 += __shfl_xor(a0, off, 32);
            a1 += __shfl_xor(a1, off, 32);
            a2 += __shfl_xor(a2, off, 32);
        }

        if (lane == 0) {
            const float* git = gi + (size_t)t * G3;
            __builtin_prefetch(git + (size_t)8 * G3 + u, 0, 0);  // 8 steps ahead
            const float r = 1.f / (1.f + __expf(-(git[u] + a0 + bh_r)));
            const float z = 1.f / (1.f + __expf(-(git[u + H_HID] + a1 + bh_z)));
            const float n = tanhf(git[u + 2 * H_HID] + r * (a2 + bh_n));
            hnew = (1.f - z) * n + z * hlds[u];
            hnxt[u] = hnew;
            __threadfence();
        }
        grid_barrier(cnt, gen, gridDim.x);
    }

    if (lane == 0) out[u] = hnew;
}

// ---------------------------------------------------------------------------
extern "C" void kernel_launch(void* const* d_in, const int* in_sizes, int n_in,
                              void* d_out, int out_size, void* d_ws, size_t ws_size,
                              hipStream_t stream) {
    (void)in_sizes; (void)n_in; (void)out_size; (void)ws_size;

    const float* src  = (const float*)d_in[0];   // [4096, 2048]
    const float* h0   = (const float*)d_in[1];   // [2048]
    const float* w_ih = (const float*)d_in[2];   // [6144, 2048]
    const float* w_hh = (const float*)d_in[3];   // [6144, 2048]
    const float* b_ih = (const float*)d_in[4];   // [6144]
    const float* b_hh = (const float*)d_in[5];   // [6144]
    float* out = (float*)d_out;                  // [2048]

    float*    gi    = (float*)d_ws;
    float*    hbufs = (float*)((char*)d_ws + HBUF_OFF);
    unsigned* bar   = (unsigned*)((char*)d_ws + BAR_OFF);

    // Re-initialize barrier + h buffers every call (graph-replay safe).
    gru_init_kernel<<<dim3((H_HID + 255) / 256), dim3(256), 0, stream>>>(h0, hbufs, bar);

    // Phase 1: WMMA GEMM. Grid: N/512 x M/16 blocks of 256 threads.
    gru_gemm_wmma<<<dim3(G3 / 512, L_SEQ / 16), dim3(256), 0, stream>>>(src, w_ih, b_ih, gi);

    // Phase 2: persistent scan, 256 WGs, 204,800 B dynamic LDS each.
    const size_t smem_bytes = (size_t)(24 * 2048 + 2048) * sizeof(float);
    gru_scan<<<dim3(H_HID / 8), dim3(256), smem_bytes, stream>>>(gi, w_hh, b_hh,
                                                                 hbufs, bar, out);
}